// MLSM_42554535969495
// MI455X (gfx1250) — compile-verified
//
#include <hip/hip_runtime.h>

typedef __attribute__((ext_vector_type(2))) float v2f;
typedef __attribute__((ext_vector_type(8))) float v8f;

#define BATCH   8
#define RNUM    10000
#define KGRP    5
#define NBLK    2000
#define FC_K    2000
#define MLP_K   10000
#define MLP_N   100
#define KLEN    200          // split-K chunk (multiple of 4!)
#define FC_SPLIT   (FC_K / KLEN)    // 10
#define MLP_SPLIT  (MLP_K / KLEN)   // 50
#define CUR0    0.05f
#define THR0    0.1f

// ---------------------------------------------------------------------------
// Pack the five (8,400) inputs into x (8,2000), row-major.
// ---------------------------------------------------------------------------
__global__ void mlsm_pack_x(const float* __restrict__ i0, const float* __restrict__ i1,
                            const float* __restrict__ i2, const float* __restrict__ i3,
                            const float* __restrict__ i4, float* __restrict__ xp) {
    int idx = blockIdx.x * blockDim.x + threadIdx.x;
    if (idx >= BATCH * FC_K) return;
    int b = idx / FC_K;
    int k = idx - b * FC_K;
    int part = k / 400;
    int off  = k - part * 400;
    const float* s = (part == 0) ? i0 : (part == 1) ? i1 : (part == 2) ? i2
                   : (part == 3) ? i3 : i4;
    xp[idx] = s[b * 400 + off];
}

// ---------------------------------------------------------------------------
// Split-K WMMA GEMM:  partial[s][m][n] = sum_{k in chunk s} A[m][k] * W[n][k]
// A: (8, Kdim) row-major (M padded to 16 by clamping rows 8..15 onto 0..7,
//    those D rows are never stored). W: (N, Kdim) row-major.
// One wave32 per (16-column tile, K-chunk); v_wmma_f32_16x16x4_f32 steps K by 4.
// VGPR layouts per CDNA5 ISA §7.12.2: lanes 16-31 carry K=2,3 (hi half).
// ---------------------------------------------------------------------------
__global__ void mlsm_wmma_gemm_splitk(const float* __restrict__ A,
                                      const float* __restrict__ W,
                                      float* __restrict__ partial,
                                      int Kdim, int N, int ntiles) {
    const int lane = threadIdx.x & 31;
    const int wv   = threadIdx.x >> 5;
    const int id   = blockIdx.x * (blockDim.x >> 5) + wv;   // wave-uniform
    const int nsplit = Kdim / KLEN;
    if (id >= ntiles * nsplit) return;        // EXEC stays all-1s (uniform)

    const int tile  = id % ntiles;
    const int split = id / ntiles;
    const int n0    = tile * 16;
    const int kbeg  = split * KLEN;

    const int lo   = lane & 15;
    const int hi   = lane >> 4;               // 0: K=0,1   1: K=2,3
    const int col  = n0 + lo;
    const int ccol = (col < N) ? col : (N - 1);   // clamp; garbage cols unstored

    const float* __restrict__ arow = A + (size_t)(lo & 7) * Kdim + 2 * hi;
    const float* __restrict__ wrow = W + (size_t)ccol   * Kdim + 2 * hi;

    v8f c = {};
#pragma unroll 4
    for (int k0 = kbeg; k0 < kbeg + KLEN; k0 += 4) {
        v2f a  = *(const v2f*)(arow + k0);    // {A[m][k0+2hi], A[m][k0+2hi+1]}
        v2f bf = *(const v2f*)(wrow + k0);    // {W[col][k0+2hi], W[col][k0+2hi+1]}
        c = __builtin_amdgcn_wmma_f32_16x16x4_f32(
                /*neg_a=*/false, a, /*neg_b=*/false, bf,
                /*c_mod=*/(short)0, c, /*reuse_a=*/false, /*reuse_b=*/false);
    }

    if (hi == 0 && col < N) {                 // lanes 0..15 hold D rows 0..7
#pragma unroll
        for (int mm = 0; mm < BATCH; ++mm)
            partial[((size_t)split * BATCH + mm) * N + col] = c[mm];
    }
}

// ---------------------------------------------------------------------------
// out[m][n] = sum_s partial[s][m][n] + bias1[n] (+ bias2[n]) + addc
// Fixed summation order -> deterministic.
// ---------------------------------------------------------------------------
__global__ void mlsm_reduce(const float* __restrict__ partial, int nsplit,
                            const float* __restrict__ bias1,
                            const float* __restrict__ bias2,
                            float addc, float* __restrict__ out, int N) {
    int idx = blockIdx.x * blockDim.x + threadIdx.x;
    if (idx >= BATCH * N) return;
    int m = idx / N;
    int col = idx - m * N;
    float s = addc;
    if (bias1) s += bias1[col];
    if (bias2) s += bias2[col];
    for (int t = 0; t < nsplit; ++t)
        s += partial[((size_t)t * BATCH + m) * N + col];
    out[idx] = s;
}

// ---------------------------------------------------------------------------
// Exact 160th/161st-largest per (batch, group): order-preserving uint keys +
// greedy MSB binary search (max u with count(key >= u) >= k == kth largest).
// One block per (b,g): 40 blocks total.
// ---------------------------------------------------------------------------
__global__ void mlsm_select_nps(const float* __restrict__ mem, float* __restrict__ nps) {
    __shared__ unsigned keys[NBLK];
    __shared__ int cnt;
    const int b = blockIdx.x / KGRP;
    const int g = blockIdx.x - b * KGRP;
    const float* base = mem + (size_t)b * RNUM + g;

    for (int i = threadIdx.x; i < NBLK; i += blockDim.x) {
        union { float f; unsigned u; } v;
        v.f = base[(size_t)i * KGRP];         // reshape(B,NB,K): r = nb*5 + g
        unsigned u = v.u;
        keys[i] = (u & 0x80000000u) ? ~u : (u | 0x80000000u);
    }
    __syncthreads();

    float q[2];
    for (int sel = 0; sel < 2; ++sel) {
        const int kth = 160 + sel;            // sel0: 160th largest, sel1: 161st
        unsigned prefix = 0u;
        for (int bit = 31; bit >= 0; --bit) {
            unsigned cand = prefix | (1u << bit);
            if (threadIdx.x == 0) cnt = 0;
            __syncthreads();
            int local = 0;
            for (int i = threadIdx.x; i < NBLK; i += blockDim.x)
                local += (keys[i] >= cand) ? 1 : 0;
            if (local) atomicAdd(&cnt, local);
            __syncthreads();
            int total = cnt;
            __syncthreads();                  // protect cnt reset next iter
            if (total >= kth) prefix = cand;
        }
        union { float f; unsigned u; } r;
        r.u = (prefix & 0x80000000u) ? (prefix ^ 0x80000000u) : ~prefix;
        q[sel] = r.f;
    }
    if (threadIdx.x == 0) {
        float q2 = q[0];                      // topq[...,159]
        float q1 = q[1];                      // topq[...,160]
        nps[blockIdx.x] = q1 + (q2 - q1) * 0.2f;
    }
}

// ---------------------------------------------------------------------------
// spike = (relu(mem - nps) - THR > 0)  ==  (mem - nps > THR)
// ---------------------------------------------------------------------------
__global__ void mlsm_spike(const float* __restrict__ mem, const float* __restrict__ nps,
                           float* __restrict__ spike) {
    int idx = blockIdx.x * blockDim.x + threadIdx.x;
    if (idx >= BATCH * RNUM) return;
    int b = idx / RNUM;
    int r = idx - b * RNUM;
    int g = r % KGRP;
    float m = mem[idx] - nps[b * KGRP + g];
    spike[idx] = (m > THR0) ? 1.0f : 0.0f;
}

// ---------------------------------------------------------------------------
extern "C" void kernel_launch(void* const* d_in, const int* in_sizes, int n_in,
                              void* d_out, int out_size, void* d_ws, size_t ws_size,
                              hipStream_t stream) {
    const float* in1   = (const float*)d_in[0];
    const float* in2   = (const float*)d_in[1];
    const float* in3   = (const float*)d_in[2];
    const float* in4   = (const float*)d_in[3];
    const float* in5   = (const float*)d_in[4];
    const float* W_fc  = (const float*)d_in[5];
    const float* b_fc  = (const float*)d_in[6];
    /* d_in[7] = W_fv: multiplied by zero spikes -> never read (saves 400 MB) */
    const float* b_fv  = (const float*)d_in[8];
    const float* W_mlp = (const float*)d_in[9];
    const float* b_mlp = (const float*)d_in[10];

    float* out   = (float*)d_out;        // (8,100) flat = 800
    float* spike = out + 800;            // (8,10000) flat = 80000

    char*  ws      = (char*)d_ws;
    float* xp      = (float*)ws;                              //  64,000 B
    float* partial = (float*)(ws + 65536);                    // 3,200,000 B (FC)
                                                              // reused for MLP (160,000 B)
    float* mem     = (float*)(ws + 65536 + 3200000);          //   320,000 B
    float* nps     = (float*)(ws + 65536 + 3200000 + 320000); //       160 B

    // 1) pack x = concat(in1..in5)  -> (8,2000)
    mlsm_pack_x<<<(BATCH * FC_K + 255) / 256, 256, 0, stream>>>(in1, in2, in3, in4, in5, xp);

    // 2a) split-K GEMM: 625 tiles x 10 K-chunks = 6250 waves (HBM-bound, 80MB)
    {
        int waves = (RNUM / 16) * FC_SPLIT;               // 6250
        int grid  = (waves + 7) / 8;
        mlsm_wmma_gemm_splitk<<<grid, 256, 0, stream>>>(xp, W_fc, partial,
                                                        FC_K, RNUM, RNUM / 16);
    }
    // 2b) reduce partials + b_fc + b_fv + 0.05 -> mem
    mlsm_reduce<<<(BATCH * RNUM + 255) / 256, 256, 0, stream>>>(
        partial, FC_SPLIT, b_fc, b_fv, CUR0, mem, RNUM);

    // 3) per-(b,g) order statistics -> nps (40 groups)
    mlsm_select_nps<<<BATCH * KGRP, 256, 0, stream>>>(mem, nps);

    // 4) spike thresholding -> second output
    mlsm_spike<<<(BATCH * RNUM + 255) / 256, 256, 0, stream>>>(mem, nps, spike);

    // 5a) split-K GEMM: 7 tiles x 50 K-chunks = 350 waves
    {
        int ntiles = (MLP_N + 15) / 16;                   // 7
        int waves  = ntiles * MLP_SPLIT;                  // 350
        int grid   = (waves + 7) / 8;
        mlsm_wmma_gemm_splitk<<<grid, 256, 0, stream>>>(spike, W_mlp, partial,
                                                        MLP_K, MLP_N, ntiles);
    }
    // 5b) reduce partials + b_mlp -> out
    mlsm_reduce<<<(BATCH * MLP_N + 255) / 256, 256, 0, stream>>>(
        partial, MLP_SPLIT, b_mlp, nullptr, 0.0f, out, MLP_N);
}